// NeuralMap_20332375179549
// MI455X (gfx1250) — compile-verified
//
#include <hip/hip_runtime.h>
#include <math.h>
#include <stdint.h>

typedef __attribute__((ext_vector_type(2))) float v2f;
typedef __attribute__((ext_vector_type(4))) float v4f;
typedef __attribute__((ext_vector_type(8))) float v8f;
typedef __attribute__((ext_vector_type(4))) unsigned int v4u;
typedef __attribute__((ext_vector_type(4))) int v4i;
typedef __attribute__((ext_vector_type(8))) int v8i;

#define UNITS 256
#define Hh 128
#define Ww 128
#define Bb 256

// CDNA5 Tensor Data Mover path (TENSORcnt), probe-guarded.
// This toolchain (clang-23 / therock-10.0) uses the 6-arg builtin form:
// (uint32x4 g0, int32x8 g1, int32x4 g2, int32x4 g3, int32x8 pad, i32 cpol)
#if defined(__AMDGCN__) && __has_builtin(__builtin_amdgcn_tensor_load_to_lds) && \
    __has_builtin(__builtin_amdgcn_s_wait_tensorcnt)
#define HAVE_TDM 1
#else
#define HAVE_TDM 0
#endif

// ---------------- workspace layout (floats) ----------------
#define OFF_CONV    0u            // 126*126*4 = 63504
#define OFF_POOLED  65536u        // 62500
#define OFF_RT      131072u       // 256
#define OFF_SCAL    131328u       // [0]=at_sumsq [1]=mem_sumsq
#define OFF_ROWSUM  131584u       // 256
#define OFF_RATIO   131840u       // 256
#define OFF_ACAT    132096u       // 256*512
#define OFF_QT      263168u       // 256*256
#define OFF_ST      328704u
#define OFF_CT      394240u
#define OFF_MEMT    459776u
#define OFF_DELTA   525312u
#define OFF_G       590848u
#define OFF_MEMNEW  656384u
#define OFF_AT      721920u       // 256*16384

#define LDSB_STRIDE 260   // 1040B row pitch: 16B-aligned rows, 260%64=4 -> conflict-free

__device__ inline v8f wmma_f32(v2f a, v2f b, v8f c) {
  return __builtin_amdgcn_wmma_f32_16x16x4_f32(false, a, false, b, (short)0, c,
                                               false, false);
}

// ---------------- init (atomic accumulators must be zeroed each launch) ----
__global__ void k_init(float* rt, float* scal) {
  int t = threadIdx.x;
  rt[t] = 0.f;
  if (t < 2) scal[t] = 0.f;
}

// ---------------- 3x3 valid conv, 256 in -> 4 out --------------------------
__global__ void k_conv(const float* __restrict__ nm, const float* __restrict__ kk,
                       float* __restrict__ conv) {
  __shared__ float red[256 * 4];
  int hw = blockIdx.x;          // 0..126*126-1
  int h = hw / 126, w = hw % 126;
  int u = threadIdx.x;
  float a0 = 0.f, a1 = 0.f, a2 = 0.f, a3 = 0.f;
  for (int dy = 0; dy < 3; ++dy)
    for (int dx = 0; dx < 3; ++dx) {
      float v = nm[((h + dy) * Ww + (w + dx)) * UNITS + u];
      const float* kp = kk + ((dy * 3 + dx) * UNITS + u) * 4;
      a0 += v * kp[0]; a1 += v * kp[1]; a2 += v * kp[2]; a3 += v * kp[3];
    }
  red[u * 4 + 0] = a0; red[u * 4 + 1] = a1;
  red[u * 4 + 2] = a2; red[u * 4 + 3] = a3;
  __syncthreads();
  for (int s = 128; s > 0; s >>= 1) {
    if (u < s) {
      red[u * 4 + 0] += red[(u + s) * 4 + 0];
      red[u * 4 + 1] += red[(u + s) * 4 + 1];
      red[u * 4 + 2] += red[(u + s) * 4 + 2];
      red[u * 4 + 3] += red[(u + s) * 4 + 3];
    }
    __syncthreads();
  }
  if (u < 4) conv[hw * 4 + u] = red[u];
}

// ---------------- 2x2 sum pool * 0.25 --------------------------------------
__global__ void k_pool(const float* __restrict__ conv, float* __restrict__ pooled) {
  int i = blockIdx.x * blockDim.x + threadIdx.x;
  if (i >= 125 * 125 * 4) return;
  int f = i & 3, hw = i >> 2, h = hw / 125, w = hw % 125;
  float p = conv[(h * 126 + w) * 4 + f] + conv[(h * 126 + w + 1) * 4 + f] +
            conv[((h + 1) * 126 + w) * 4 + f] + conv[((h + 1) * 126 + w + 1) * 4 + f];
  pooled[i] = 0.25f * p;
}

// ---------------- r_t = pooled @ conv_dense1 (64MB stream, NT + prefetch) --
__global__ void k_dense(const float* __restrict__ pooled, const float* __restrict__ W,
                        float* __restrict__ rt) {
  int j = threadIdx.x;
  int r0 = blockIdx.x * 250;
  float acc = 0.f;
  for (int i = r0; i < r0 + 250; ++i) {
    __builtin_prefetch(&W[(i + 16) * 256 + j], 0, 0);   // global_prefetch_b8
    acc += pooled[i] * __builtin_nontemporal_load(&W[i * 256 + j]);
  }
  atomicAdd(&rt[j], acc);
}

// ---------------- Acat = [context | broadcast(r_t)] ------------------------
__global__ void k_acat(const float* __restrict__ ctx, const float* __restrict__ rt,
                       float* __restrict__ A) {
  int i = blockIdx.x * 256 + threadIdx.x;    // 0..131071
  int b = i >> 9, k = i & 511;
  A[i] = (k < 256) ? ctx[b * 256 + k] : rt[k - 256];
}

// ---------------- generic row-major WMMA f32 GEMM, 1 wave per 16x16 tile ---
__global__ void k_gemm_rr(const float* __restrict__ A, int lda,
                          const float* __restrict__ B, int ldb,
                          float* __restrict__ D, int ldd, int K) {
  int m0 = blockIdx.y * 16, n0 = blockIdx.x * 16;
  int lane = threadIdx.x & 31, l = lane & 15, hf = lane >> 4;
  v8f c = {};
  for (int k0 = 0; k0 < K; k0 += 4) {
    v2f a, b;
    a.x = A[(m0 + l) * lda + k0 + 2 * hf];
    a.y = A[(m0 + l) * lda + k0 + 2 * hf + 1];
    b.x = B[(k0 + 2 * hf) * ldb + n0 + l];
    b.y = B[(k0 + 2 * hf + 1) * ldb + n0 + l];
    c = wmma_f32(a, b, c);
  }
  for (int v = 0; v < 8; ++v)
    D[(m0 + v + 8 * hf) * ldd + n0 + l] = c[v];
}

// ---------------- at[b][p] = sum_u qt[b][u]*nm[p][u]  (B transposed via LDS)
// LDS tile filled by the Tensor Data Mover: one 2D-tile descriptor per block,
// hardware pad feature (interval=256 DW, amount=4 DW) creates the 260-float
// conflict-free row pitch. Fused frobenius sum-of-squares accumulation.
__global__ void k_gemm_at(const float* __restrict__ qt, const float* __restrict__ nm,
                          float* __restrict__ at, float* __restrict__ norm_slot) {
  __shared__ float ldsB[16 * LDSB_STRIDE];
  __shared__ float red[256];
  int tid = threadIdx.x;
  int p0 = blockIdx.x * 16;          // n tile (p)
  int mb = blockIdx.y;               // 0..1
#if HAVE_TDM
  if (tid < 32) {                    // one wave issues the DMA (EXEC ignored by TDM)
    unsigned lds_base = (unsigned)(uintptr_t)(&ldsB[0]);
    unsigned long long ga = (unsigned long long)(uintptr_t)(nm + p0 * UNITS);
    // D# group 0: count=1 | lds_addr | global_addr | type=2
    v4u g0 = {1u, lds_base, (unsigned)(ga & 0xFFFFFFFFu),
              ((unsigned)((ga >> 32) & 0x01FFFFFFu)) | (2u << 30)};
    // D# group 1: data_size=4B, pad_enable, pad_interval=256DW, pad_amount=4DW,
    // tensor_dim0=256, tensor_dim1=16384, tile_dim0=256, tile_dim1=16,
    // tensor_dim0_stride=256
    v8i g1 = {(int)((2u << 16) | (1u << 20) | (7u << 22) | (3u << 25)),
              (int)(256u << 16),     // tensor_dim0[15:0] @ [63:48]
              (int)(16384u << 16),   // tensor_dim1[15:0] @ [95:80]
              (int)(256u << 16),     // tile_dim0 @ [127:112]
              16,                    // tile_dim1 @ [143:128]
              256,                   // tensor_dim0_stride[31:0]
              0, 0};
    v4i gz4 = {0, 0, 0, 0};
    v8i gz8 = {0, 0, 0, 0, 0, 0, 0, 0};
    __builtin_amdgcn_tensor_load_to_lds(g0, g1, gz4, gz4, gz8, 0);
    __builtin_amdgcn_s_wait_tensorcnt(0);
  }
  __syncthreads();
#else
  for (int i = 0; i < 16; ++i)
    ldsB[i * LDSB_STRIDE + tid] = nm[(p0 + i) * UNITS + tid];
  __syncthreads();
#endif
  int w = tid >> 5, lane = tid & 31, l = lane & 15, hf = lane >> 4;
  int m0 = mb * 128 + w * 16;
  v8f c = {};
  for (int k0 = 0; k0 < 256; k0 += 4) {
    v2f a, b;
    a.x = qt[(m0 + l) * 256 + k0 + 2 * hf];
    a.y = qt[(m0 + l) * 256 + k0 + 2 * hf + 1];
    b.x = ldsB[l * LDSB_STRIDE + k0 + 2 * hf];
    b.y = ldsB[l * LDSB_STRIDE + k0 + 2 * hf + 1];
    c = wmma_f32(a, b, c);
  }
  float ss = 0.f;
  for (int v = 0; v < 8; ++v) {
    float val = c[v];
    at[(m0 + v + 8 * hf) * 16384 + p0 + l] = val;
    ss += val * val;
  }
  red[tid] = ss;
  __syncthreads();
  for (int s = 128; s > 0; s >>= 1) {
    if (tid < s) red[tid] += red[tid + s];
    __syncthreads();
  }
  if (tid == 0) atomicAdd(norm_slot, red[0]);
}

// ---------------- at = exp(at/||at||); rowsum[b] ---------------------------
__global__ void k_exp_rowsum(float* __restrict__ at, const float* __restrict__ scal,
                             float* __restrict__ rowsum) {
  __shared__ float red[256];
  int b = blockIdx.x, tid = threadIdx.x;
  float scale = rsqrtf(scal[0]);
  float* row = at + b * 16384;
  float s = 0.f;
  for (int i = tid; i < 16384; i += 256) {
    float e = __expf(row[i] * scale);
    row[i] = e;
    s += e;
  }
  red[tid] = s;
  __syncthreads();
  for (int st = 128; st > 0; st >>= 1) {
    if (tid < st) red[tid] += red[tid + st];
    __syncthreads();
  }
  if (tid == 0) rowsum[b] = red[0];
}

// ---------------- c_t = (at @ nm) / rowsum -----------------------------------
// 16x32 tile per wave (A-fragment reused across 2 WMMAs), split-K over 8 waves
__global__ void k_gemm_ct(const float* __restrict__ at, const float* __restrict__ nm,
                          const float* __restrict__ rowsum, float* __restrict__ ct) {
  __shared__ float part[8 * 512];
  int tid = threadIdx.x;
  int w = tid >> 5, lane = tid & 31, l = lane & 15, hf = lane >> 4;
  int m0 = blockIdx.y * 16, n0 = blockIdx.x * 32;
  int kbeg = w * 2048, kend = kbeg + 2048;
  v8f c0 = {}, c1 = {};
  for (int k0 = kbeg; k0 < kend; k0 += 4) {
    v2f a, b0, b1;
    a.x = at[(m0 + l) * 16384 + k0 + 2 * hf];
    a.y = at[(m0 + l) * 16384 + k0 + 2 * hf + 1];
    const float* Brow0 = nm + (k0 + 2 * hf) * 256 + n0;
    const float* Brow1 = Brow0 + 256;
    b0.x = Brow0[l];      b0.y = Brow1[l];
    b1.x = Brow0[16 + l]; b1.y = Brow1[16 + l];
    c0 = wmma_f32(a, b0, c0);
    c1 = wmma_f32(a, b1, c1);
  }
  for (int v = 0; v < 8; ++v) {
    int r = v + 8 * hf;
    part[w * 512 + r * 32 + l]      = c0[v];
    part[w * 512 + r * 32 + 16 + l] = c1[v];
  }
  __syncthreads();
  for (int e = tid; e < 512; e += 256) {
    float s = 0.f;
    for (int ww = 0; ww < 8; ++ww) s += part[ww * 512 + e];
    int r = e >> 5, cc = e & 31;
    ct[(m0 + r) * 256 + n0 + cc] = s / rowsum[m0 + r];
  }
}

// ---------------- ratio[b] = local/(local+global), with 0 -> +1 fix --------
__global__ void k_imp(const float* __restrict__ stv, const float* __restrict__ rt,
                      const float* __restrict__ ct, float* __restrict__ ratio) {
  __shared__ float redL[256];
  __shared__ float redG[256];
  int b = blockIdx.x, u = threadIdx.x;
  float s = stv[b * 256 + u];
  redG[u] = s * rt[u];
  redL[u] = s * ct[b * 256 + u];
  __syncthreads();
  for (int st = 128; st > 0; st >>= 1) {
    if (u < st) { redL[u] += redL[u + st]; redG[u] += redG[u + st]; }
    __syncthreads();
  }
  if (u == 0) {
    float L = redL[0], G = redG[0], comb = L + G;
    if (comb == 0.f) comb += 1.f;
    ratio[b] = L / comb;
  }
}

// ---------------- gather mem_t rows; delta = mem_t - s_t -------------------
__global__ void k_delta(const float* __restrict__ nm, const int* __restrict__ pos,
                        const float* __restrict__ stv, float* __restrict__ memt,
                        float* __restrict__ delta) {
  int b = blockIdx.x, u = threadIdx.x;
  int x = pos[b * 2 + 0], y = pos[b * 2 + 1];
  float m = nm[(y * Ww + x) * UNITS + u];
  memt[b * 256 + u] = m;
  delta[b * 256 + u] = m - stv[b * 256 + u];
}

// ---------------- mem_new = mem_t + ratio*G; accumulate sumsq --------------
__global__ void k_memnew(const float* __restrict__ memt, const float* __restrict__ G,
                         const float* __restrict__ ratio, float* __restrict__ memnew,
                         float* __restrict__ sumsq) {
  __shared__ float red[256];
  int b = blockIdx.x, u = threadIdx.x;
  float v = memt[b * 256 + u] + ratio[b] * G[b * 256 + u];
  memnew[b * 256 + u] = v;
  red[u] = v * v;
  __syncthreads();
  for (int s = 128; s > 0; s >>= 1) {
    if (u < s) red[u] += red[u + s];
    __syncthreads();
  }
  if (u == 0) atomicAdd(sumsq, red[0]);
}

// ---------------- copy nm -> out_map (16.7MB, NT both ways) ----------------
__global__ void k_copy(const v4f* __restrict__ src, v4f* __restrict__ dst, int n4) {
  int i = blockIdx.x * blockDim.x + threadIdx.x;
  if (i < n4) {
    v4f v = __builtin_nontemporal_load(&src[i]);
    __builtin_nontemporal_store(v, &dst[i]);
  }
}

// ---------------- out1 = c_t + mem_norm; last-writer-wins scatter ----------
__global__ void k_final(const float* __restrict__ memnew, const float* __restrict__ ct,
                        const float* __restrict__ scal, const int* __restrict__ pos,
                        float* __restrict__ out1, float* __restrict__ omap) {
  __shared__ int win;
  int b = blockIdx.x, u = threadIdx.x;
  if (u == 0) win = 1;
  __syncthreads();
  int x = pos[b * 2 + 0], y = pos[b * 2 + 1];
  if (u > b && pos[u * 2] == x && pos[u * 2 + 1] == y) win = 0;  // benign: all write 0
  __syncthreads();
  float scale = rsqrtf(scal[1]);
  float mn = memnew[b * 256 + u] * scale;
  out1[b * 256 + u] = ct[b * 256 + u] + mn;
  if (win) omap[(y * Ww + x) * UNITS + u] = mn;
}

extern "C" void kernel_launch(void* const* d_in, const int* in_sizes, int n_in,
                              void* d_out, int out_size, void* d_ws, size_t ws_size,
                              hipStream_t stream) {
  (void)in_sizes; (void)n_in; (void)out_size; (void)ws_size;
  const float* ctx = (const float*)d_in[0];
  const int*   pos = (const int*)d_in[1];
  const float* nm  = (const float*)d_in[2];
  const float* ck1 = (const float*)d_in[3];
  const float* cd1 = (const float*)d_in[4];
  const float* ctk = (const float*)d_in[5];
  const float* wk  = (const float*)d_in[6];
  const float* wu  = (const float*)d_in[7];

  float* ws      = (float*)d_ws;
  float* conv    = ws + OFF_CONV;
  float* pooled  = ws + OFF_POOLED;
  float* rt      = ws + OFF_RT;
  float* scal    = ws + OFF_SCAL;
  float* rowsum  = ws + OFF_ROWSUM;
  float* ratio   = ws + OFF_RATIO;
  float* acat    = ws + OFF_ACAT;
  float* qt      = ws + OFF_QT;
  float* stv     = ws + OFF_ST;
  float* ct      = ws + OFF_CT;
  float* memt    = ws + OFF_MEMT;
  float* delta   = ws + OFF_DELTA;
  float* G       = ws + OFF_G;
  float* memnew  = ws + OFF_MEMNEW;
  float* at      = ws + OFF_AT;

  float* out1 = (float*)d_out;                 // c_t + normalized mem_new (256x256)
  float* omap = (float*)d_out + 256 * 256;     // new_map (1,128,128,256)

  k_init<<<1, 256, 0, stream>>>(rt, scal);
  k_conv<<<126 * 126, 256, 0, stream>>>(nm, ck1, conv);
  k_pool<<<(125 * 125 * 4 + 255) / 256, 256, 0, stream>>>(conv, pooled);
  k_dense<<<250, 256, 0, stream>>>(pooled, cd1, rt);
  k_acat<<<512, 256, 0, stream>>>(ctx, rt, acat);
  k_gemm_rr<<<dim3(16, 16), 32, 0, stream>>>(acat, 512, ctk, 256, qt, 256, 512);
  k_gemm_rr<<<dim3(16, 16), 32, 0, stream>>>(ctx, 256, wk, 256, stv, 256, 256);
  k_gemm_at<<<dim3(1024, 2), 256, 0, stream>>>(qt, nm, at, scal);
  k_exp_rowsum<<<256, 256, 0, stream>>>(at, scal, rowsum);
  k_gemm_ct<<<dim3(8, 16), 256, 0, stream>>>(at, nm, rowsum, ct);
  k_imp<<<256, 256, 0, stream>>>(stv, rt, ct, ratio);
  k_delta<<<256, 256, 0, stream>>>(nm, pos, stv, memt, delta);
  k_gemm_rr<<<dim3(16, 16), 32, 0, stream>>>(delta, 256, wu, 256, G, 256, 256);
  k_memnew<<<256, 256, 0, stream>>>(memt, G, ratio, memnew, scal + 1);
  k_copy<<<4096, 256, 0, stream>>>((const v4f*)nm, (v4f*)omap, 1048576);
  k_final<<<256, 256, 0, stream>>>(memnew, ct, scal, pos, out1, omap);
}